// MultiHeadAttention_71889162600746
// MI455X (gfx1250) — compile-verified
//
#include <hip/hip_runtime.h>

typedef __attribute__((ext_vector_type(2))) float v2f;
typedef __attribute__((ext_vector_type(8))) float v8f;

#define NEG_SLOPE 0.2f
#define DIM 128

__device__ __forceinline__ float leaky(float v) { return v >= 0.f ? v : NEG_SLOPE * v; }

// Monotone order-preserving float->uint mapping so segment-max lowers to
// native global_atomic_max_u32 (no CAS loop).
__device__ __forceinline__ unsigned enc_f32(float f) {
    unsigned b = __float_as_uint(f);
    return (b & 0x80000000u) ? ~b : (b | 0x80000000u);
}
__device__ __forceinline__ float dec_f32(unsigned u) {
    unsigned b = (u & 0x80000000u) ? (u ^ 0x80000000u) : ~u;
    return __uint_as_float(b);
}
// enc(-inf) == 0x007FFFFF
#define ENC_NEG_INF 0x007FFFFFu

// ---------------------------------------------------------------- init
__global__ void init_kernel(float* __restrict__ out, float* __restrict__ qagg,
                            unsigned* __restrict__ mx, float* __restrict__ sm, int N) {
    int i = blockIdx.x * blockDim.x + threadIdx.x;
    if (i < N * DIM) out[i] = 0.f;
    if (i < N * 4) { qagg[i] = 0.f; sm[i] = 0.f; mx[i] = ENC_NEG_INF; }
}

// ------------------------------------------- per-node scalar projections
// hq[n][h] = x[n] . Wq[h] ;  kact[n][h] = leaky(x[n] . Wk[h])
__global__ __launch_bounds__(256) void small_proj_kernel(
    const float* __restrict__ x, const float* __restrict__ Wq,
    const float* __restrict__ Wk, float* __restrict__ hq,
    float* __restrict__ kact, int N) {
    __shared__ float wq[4 * DIM];
    __shared__ float wk[4 * DIM];
    for (int i = threadIdx.x; i < 4 * DIM; i += 256) { wq[i] = Wq[i]; wk[i] = Wk[i]; }
    __syncthreads();
    int n = blockIdx.x * 256 + threadIdx.x;
    if (n >= N) return;
    const float4* xr = (const float4*)(x + (size_t)n * DIM);
    float aq[4] = {0.f, 0.f, 0.f, 0.f};
    float ak[4] = {0.f, 0.f, 0.f, 0.f};
    for (int k4 = 0; k4 < DIM / 4; ++k4) {
        float4 xv = xr[k4];
#pragma unroll
        for (int h = 0; h < 4; ++h) {
            float4 qv = ((const float4*)(wq + h * DIM))[k4];
            float4 kv = ((const float4*)(wk + h * DIM))[k4];
            aq[h] += xv.x * qv.x + xv.y * qv.y + xv.z * qv.z + xv.w * qv.w;
            ak[h] += xv.x * kv.x + xv.y * kv.y + xv.z * kv.z + xv.w * kv.w;
        }
    }
#pragma unroll
    for (int h = 0; h < 4; ++h) {
        hq[n * 4 + h] = aq[h];
        kact[n * 4 + h] = leaky(ak[h]);
    }
}

// ----------------------------------- h_proj = x @ Wl.T + bl  (fp32 WMMA)
// Block: 256 threads = 8 waves (wave32). M-block = 64 rows. Each wave owns a
// 16-wide N-tile; LDS holds Wl transposed (k-major) + the 64x128 x tile
// (row padded to 132 floats to avoid 16-way bank conflicts on A-frag reads).
#define M_BLK 64
#define A_PITCH 132
__global__ __launch_bounds__(256) void hproj_wmma_kernel(
    const float* __restrict__ x, const float* __restrict__ Wl,
    const float* __restrict__ bl, float* __restrict__ hproj, int N) {
    extern __shared__ float lds[];
    float* bt = lds;                  // [128][128]  bt[k*128+n] = Wl[n*128+k]
    float* at = lds + DIM * DIM;      // [64][132]   x tile
    const int t = threadIdx.x;
    const int node0 = blockIdx.x * M_BLK;

    for (int i = t; i < DIM * DIM; i += 256) {
        int n = i >> 7, k = i & (DIM - 1);
        bt[k * DIM + n] = Wl[i];                 // coalesced read, transposed write
    }
    for (int i = t; i < M_BLK * DIM; i += 256) {
        int m = i >> 7, k = i & (DIM - 1);
        int r = node0 + m; if (r >= N) r = N - 1;
        at[m * A_PITCH + k] = x[(size_t)r * DIM + k];
    }
    __syncthreads();

    const int wave = t >> 5;          // 0..7 -> N-tile
    const int lane = t & 31;
    const int l    = lane & 15;
    const int kh   = (lane >> 4) << 1;   // 0 or 2 (K-half select)
    const int n0   = wave * 16;
    const float bv = bl[n0 + l];

    for (int mt = 0; mt < M_BLK / 16; ++mt) {
        const int m0 = mt * 16;
        v8f c = {};
        for (int kk = 0; kk < DIM; kk += 4) {
            v2f a, b;
            // A 16x4 f32: VGPR0 = K {0|2}, VGPR1 = K {1|3}, lanes 0-15 are M
            a.x = at[(m0 + l) * A_PITCH + kk + kh];
            a.y = at[(m0 + l) * A_PITCH + kk + kh + 1];
            // B 4x16 f32: rows striped across lanes, K {0|2} then {1|3}
            b.x = bt[(kk + kh) * DIM + n0 + l];
            b.y = bt[(kk + kh + 1) * DIM + n0 + l];
            c = __builtin_amdgcn_wmma_f32_16x16x4_f32(
                    false, a, false, b, (short)0, c, false, false);
        }
        // C layout: VGPR r -> M = r (+8 for lanes 16-31), N = lane%16
        const int rbase = node0 + m0 + ((lane >> 4) << 3);
#pragma unroll
        for (int r = 0; r < 8; ++r) {
            int rr = rbase + r;
            if (rr < N) hproj[(size_t)rr * DIM + n0 + l] = c[r] + bv;
        }
    }
}

// ------------------------------------------------ q_agg scatter (per edge)
__global__ void qagg_kernel(const int* __restrict__ row, const int* __restrict__ col,
                            const float* __restrict__ hq, float* __restrict__ qagg, int E) {
    int e = blockIdx.x * blockDim.x + threadIdx.x;
    if (e >= E) return;
    int r = row[e], c = col[e];
    float4 v = *(const float4*)(hq + c * 4);
    atomicAdd(&qagg[r * 4 + 0], v.x);
    atomicAdd(&qagg[r * 4 + 1], v.y);
    atomicAdd(&qagg[r * 4 + 2], v.z);
    atomicAdd(&qagg[r * 4 + 3], v.w);
}

// ------------------------------- scores + segment max (encoded atomic umax)
__global__ void scores_kernel(const int* __restrict__ row, const int* __restrict__ col,
                              const float* __restrict__ kact, const float* __restrict__ qagg,
                              float* __restrict__ scores, unsigned* __restrict__ mx, int E) {
    int e = blockIdx.x * blockDim.x + threadIdx.x;
    if (e >= E) return;
    int r = row[e], c = col[e];
    float4 ka = *(const float4*)(kact + r * 4);
    float4 qa = *(const float4*)(qagg + c * 4);
    float s0 = ka.x * qa.x, s1 = ka.y * qa.y, s2 = ka.z * qa.z, s3 = ka.w * qa.w;
    float4* sp = (float4*)(scores + (size_t)e * 4);
    *sp = make_float4(s0, s1, s2, s3);
    atomicMax(&mx[r * 4 + 0], enc_f32(s0));
    atomicMax(&mx[r * 4 + 1], enc_f32(s1));
    atomicMax(&mx[r * 4 + 2], enc_f32(s2));
    atomicMax(&mx[r * 4 + 3], enc_f32(s3));
}

// -------------------------------------------------- exp + segment sum
__global__ void expsum_kernel(const int* __restrict__ row, float* __restrict__ scores,
                              const unsigned* __restrict__ mx, float* __restrict__ sm, int E) {
    int e = blockIdx.x * blockDim.x + threadIdx.x;
    if (e >= E) return;
    int r = row[e];
    float4 s = *(const float4*)(scores + (size_t)e * 4);
    float e0 = __expf(s.x - dec_f32(mx[r * 4 + 0]));
    float e1 = __expf(s.y - dec_f32(mx[r * 4 + 1]));
    float e2 = __expf(s.z - dec_f32(mx[r * 4 + 2]));
    float e3 = __expf(s.w - dec_f32(mx[r * 4 + 3]));
    *(float4*)(scores + (size_t)e * 4) = make_float4(e0, e1, e2, e3);
    atomicAdd(&sm[r * 4 + 0], e0);
    atomicAdd(&sm[r * 4 + 1], e1);
    atomicAdd(&sm[r * 4 + 2], e2);
    atomicAdd(&sm[r * 4 + 3], e3);
}

// ---------------------------------------- alpha_mean = mean_h ex/(sum+eps)
__global__ void alpha_kernel(const int* __restrict__ row, const float* __restrict__ scores,
                             const float* __restrict__ sm, float* __restrict__ amean, int E) {
    int e = blockIdx.x * blockDim.x + threadIdx.x;
    if (e >= E) return;
    int r = row[e];
    float4 ex = *(const float4*)(scores + (size_t)e * 4);
    float a = ex.x / (sm[r * 4 + 0] + 1e-8f) + ex.y / (sm[r * 4 + 1] + 1e-8f) +
              ex.z / (sm[r * 4 + 2] + 1e-8f) + ex.w / (sm[r * 4 + 3] + 1e-8f);
    amean[e] = 0.25f * a;
}

// ------------------- wave-per-edge weighted scatter into output accumulator
__global__ __launch_bounds__(256) void scatter_kernel(
    const int* __restrict__ row, const int* __restrict__ col,
    const float* __restrict__ hproj, const float* __restrict__ amean,
    float* __restrict__ out, int E) {
    int gw = (blockIdx.x * 256 + threadIdx.x) >> 5;   // one wave32 per edge
    int lane = threadIdx.x & 31;
    if (gw >= E) return;
    if (lane == 0 && gw + 64 < E) {
        __builtin_prefetch(&row[gw + 64], 0, 0);
        __builtin_prefetch(&col[gw + 64], 0, 0);
        __builtin_prefetch(&amean[gw + 64], 0, 0);
    }
    int r = row[gw], c = col[gw];
    float am = amean[gw];
    float4 v = *(const float4*)(hproj + (size_t)c * DIM + lane * 4);
    float* o = out + (size_t)r * DIM + lane * 4;
    atomicAdd(o + 0, am * v.x);
    atomicAdd(o + 1, am * v.y);
    atomicAdd(o + 2, am * v.z);
    atomicAdd(o + 3, am * v.w);
}

// ---------------------------------------------------------- final leaky
__global__ void final_kernel(float* __restrict__ out, int total) {
    int i = blockIdx.x * blockDim.x + threadIdx.x;
    if (i < total) out[i] = leaky(out[i]);
}

extern "C" void kernel_launch(void* const* d_in, const int* in_sizes, int n_in,
                              void* d_out, int out_size, void* d_ws, size_t ws_size,
                              hipStream_t stream) {
    const float* x  = (const float*)d_in[0];
    const int*   ei = (const int*)d_in[1];
    const float* Wq = (const float*)d_in[2];
    const float* Wk = (const float*)d_in[3];
    const float* Wl = (const float*)d_in[4];
    const float* bl = (const float*)d_in[5];
    const int N = in_sizes[0] / DIM;
    const int E = in_sizes[1] / 2;
    const int* row = ei;
    const int* col = ei + E;
    float* out = (float*)d_out;

    // workspace carve-up (all fp32-sized, 16B aligned throughout)
    float*    q_agg  = (float*)d_ws;                       // N*4
    unsigned* mx     = (unsigned*)(q_agg + (size_t)N * 4); // N*4
    float*    sm     = (float*)(mx + (size_t)N * 4);       // N*4
    float*    hq     = sm + (size_t)N * 4;                 // N*4
    float*    kact   = hq + (size_t)N * 4;                 // N*4
    float*    scores = kact + (size_t)N * 4;               // E*4
    float*    amean  = scores + (size_t)E * 4;             // E
    float*    hproj  = amean + (size_t)E;                  // N*128

    const int nTot = N * DIM;
    init_kernel<<<(nTot + 255) / 256, 256, 0, stream>>>(out, q_agg, mx, sm, N);
    small_proj_kernel<<<(N + 255) / 256, 256, 0, stream>>>(x, Wq, Wk, hq, kact, N);

    const size_t ldsBytes = (size_t)(DIM * DIM + M_BLK * A_PITCH) * sizeof(float);
    hproj_wmma_kernel<<<(N + M_BLK - 1) / M_BLK, 256, ldsBytes, stream>>>(x, Wl, bl, hproj, N);

    const int eBlocks = (E + 255) / 256;
    qagg_kernel<<<eBlocks, 256, 0, stream>>>(row, col, hq, q_agg, E);
    scores_kernel<<<eBlocks, 256, 0, stream>>>(row, col, kact, q_agg, scores, mx, E);
    expsum_kernel<<<eBlocks, 256, 0, stream>>>(row, scores, mx, sm, E);
    alpha_kernel<<<eBlocks, 256, 0, stream>>>(row, scores, sm, amean, E);

    scatter_kernel<<<(E + 7) / 8, 256, 0, stream>>>(row, col, hproj, amean, out, E);
    final_kernel<<<(nTot + 255) / 256, 256, 0, stream>>>(out, nTot);
}